// ClassCapsule_62234076119301
// MI455X (gfx1250) — compile-verified
//
#include <hip/hip_runtime.h>
#include <hip/hip_bf16.h>
#include <math.h>

typedef __attribute__((ext_vector_type(2))) float v2f;
typedef __attribute__((ext_vector_type(8))) float v8f;

#define Bsz 64
#define Ssz 14
#define Isz 32
#define Csz 10
#define NPOS (Ssz*Ssz*Isz)   /* 6272 */
#define DD 16
#define CD (Csz*DD)          /* 160 */
#define SS2 (Ssz*Ssz)        /* 196 */
#define EPSf 1e-9f
#define CHUNK_N 64
#define THREADS 256

// Fused capsule EM-routing kernel: one workgroup per batch element.
// votes computed on the fly with v_wmma_f32_16x16x4_f32 (K=4 == pose dim),
// coordinate-addition folded into the WMMA C accumulator, rr recomputed per
// pass from the tiny per-class stats -> votes tensor (257MB) never touches HBM.
__global__ __launch_bounds__(THREADS)
void em_caps_kernel(const float* __restrict__ pose,
                    const float* __restrict__ act,
                    const float* __restrict__ w,
                    const float* __restrict__ beta_v,
                    const float* __restrict__ beta_a,
                    float* __restrict__ out)
{
    __shared__ float s_votes[CHUNK_N * CD];   // 40960 B
    __shared__ float s_rrp  [CHUNK_N * Csz];  //  2560 B
    __shared__ float s_mean [CD];
    __shared__ float s_inv2v[CD];
    __shared__ float s_logstd[Csz];
    __shared__ float s_logact[Csz];
    __shared__ float s_accM[CD];
    __shared__ float s_accV[CD];
    __shared__ float s_accR[Csz];
    __shared__ float s_coordH[SS2];
    __shared__ float s_coordW[SS2];

    const int b      = blockIdx.x;
    const int tid    = threadIdx.x;
    const int wave   = tid >> 5;
    const int lane   = tid & 31;
    const int laneLo = lane & 15;
    const int laneHi = lane >> 4;

    // ---- Coordinate tables: one-time, kills all /14 %14 in the hot loop ----
    if (tid < SS2) {
        s_coordH[tid] = ((float)(tid / Ssz) + 0.5f) * (1.0f / Ssz);
        s_coordW[tid] = ((float)(tid % Ssz) + 0.5f) * (1.0f / Ssz);
    }

    // ---- Preload B fragments (loop invariant): W[i,c,q,r] as B[q, c_l*4+r] ----
    // bfrag[half][t][j]: i = half*16 + wave*2 + t, classes c0 = 4*j (c>=10 padded 0)
    const int r_ln  = laneLo & 3;   // r  for this lane's B/D column
    const int cl_ln = laneLo >> 2;  // c_l for this lane's B/D column
    v2f bfrag[2][2][3];
    {
        const int q0 = laneHi * 2;
#pragma unroll
        for (int half = 0; half < 2; ++half)
#pragma unroll
            for (int t = 0; t < 2; ++t) {
                const int i = half * 16 + wave * 2 + t;
#pragma unroll
                for (int j = 0; j < 3; ++j) {
                    const int c = 4 * j + cl_ln;
                    v2f bb; bb.x = 0.0f; bb.y = 0.0f;
                    if (j < 2 || cl_ln < 2) {       // c < 10
                        const int base = ((i * Csz + c) * 4 + q0) * 4 + r_ln;
                        bb.x = w[base];             // q = q0
                        bb.y = w[base + 4];         // q = q0+1
                    }
                    bfrag[half][t][j] = bb;
                }
            }
    }
    __syncthreads();

    const int my_c = tid >> 4;   // valid when tid < 160
    const int my_d = tid & 15;

    // A-fragment lane geometry (constant per lane)
    const int a_sl = laneLo >> 2;        // spatial-local row
    const int a_p  = laneLo & 3;         // pose row p
    const int a_q0 = laneHi * 2;         // first q of this half-lane

    for (int it = 0; it < 3; ++it) {
        float accM = 0.0f, accV = 0.0f, accR = 0.0f;

        for (int sb = 0; sb < 49; ++sb) {        // 4 spatial positions per block
            const int s_base = sb * 4;

            // Coordinate values feeding cf[0] (row s2a, p=0) and cf[4] (row s2b, p=0)
            const int s2a = s_base + 2 * laneHi;
            const int s2b = s2a + 1;
            const float cA = (r_ln == 0) ? s_coordH[s2a] : ((r_ln == 1) ? s_coordW[s2a] : 0.0f);
            const float cB = (r_ln == 0) ? s_coordH[s2b] : ((r_ln == 1) ? s_coordW[s2b] : 0.0f);

#pragma unroll
            for (int half = 0; half < 2; ++half) {
                const int i_base = half * 16;

                // ---------- Phase 1: votes via WMMA ----------
                // Preload both A fragments first to overlap the global loads.
                v2f afr[2];
#pragma unroll
                for (int t = 0; t < 2; ++t) {
                    const int i  = i_base + wave * 2 + t;
                    const int pb = (((b * SS2 + (s_base + a_sl)) * 32 + i) * 16) + a_p * 4 + a_q0;
                    afr[t].x = pose[pb];
                    afr[t].y = pose[pb + 1];
                    if (sb + 1 < 49)   // prefetch next spatial block's pose
                        __builtin_prefetch(&pose[pb + 4 * 32 * 16], 0, 1);
                }

#pragma unroll
                for (int t = 0; t < 2; ++t) {
                    const int ii = wave * 2 + t;
#pragma unroll
                    for (int j = 0; j < 3; ++j) {
                        // C accumulator carries the coordinate addition; only
                        // v=0 (row p=0 of s2a) and v=4 (row p=0 of s2b) are nonzero.
                        v8f cf = {};
                        cf[0] = (j < 2 || cl_ln < 2) ? cA : 0.0f;
                        cf[4] = (j < 2 || cl_ln < 2) ? cB : 0.0f;
                        v8f dm = __builtin_amdgcn_wmma_f32_16x16x4_f32(
                            false, afr[t], false, bfrag[half][t][j],
                            (short)0, cf, false, false);
                        // write to LDS: n_local = sl2*16 + ii, d = p2*4 + r
                        const int c = 4 * j + cl_ln;
                        if (j < 2 || cl_ln < 2) {
#pragma unroll
                            for (int v = 0; v < 8; ++v) {
                                const int rw  = v + 8 * laneHi;
                                const int sl2 = rw >> 2;
                                const int p2  = rw & 3;
                                s_votes[(sl2 * 16 + ii) * CD + c * DD + p2 * 4 + r_ln] = dm[v];
                            }
                        }
                    }
                }
                __syncthreads();

                // ---------- Phase 2: rr_p[n,c] ----------
                if (tid < CHUNK_N) {
                    const int sl = tid >> 4;
                    const int ii = tid & 15;
                    const int ng = (s_base + sl) * 32 + i_base + ii;
                    const float ia = act[b * NPOS + ng];
                    const float* vrow = &s_votes[tid * CD];
                    if (it == 0) {
                        const float rp = ia * (1.0f / Csz);
#pragma unroll
                        for (int c = 0; c < Csz; ++c) s_rrp[tid * Csz + c] = rp;
                    } else {
                        float z[Csz];
                        float zmax = -1e30f;
#pragma unroll
                        for (int c = 0; c < Csz; ++c) {
                            float ssum = 0.0f;
#pragma unroll
                            for (int d = 0; d < DD; ++d) {
                                const float df = vrow[c * DD + d] - s_mean[c * DD + d];
                                ssum += df * df * s_inv2v[c * DD + d];
                            }
                            z[c] = s_logact[c] - ssum - s_logstd[c];
                            zmax = fmaxf(zmax, z[c]);
                        }
                        float esum = 0.0f;
#pragma unroll
                        for (int c = 0; c < Csz; ++c) { z[c] = __expf(z[c] - zmax); esum += z[c]; }
                        const float sc = ia / esum;
#pragma unroll
                        for (int c = 0; c < Csz; ++c) s_rrp[tid * Csz + c] = z[c] * sc;
                    }
                }
                __syncthreads();

                // ---------- Phase 3: per-(c,d) accumulation ----------
                if (tid < CD) {
                    for (int n = 0; n < CHUNK_N; ++n) {
                        const float v  = s_votes[n * CD + tid];
                        const float rp = s_rrp[n * Csz + my_c];
                        accM += rp * v;
                        accV += rp * v * v;
                        if (my_d == 0) accR += rp;
                    }
                }
                __syncthreads();
            }
        }

        // ---------- Finalize M-step ----------
        if (tid < CD) {
            s_accM[tid] = accM;
            s_accV[tid] = accV;
            if (my_d == 0) s_accR[my_c] = accR;
        }
        __syncthreads();
        if (tid < Csz) {
            const int c = tid;
            const float rsum = s_accR[c];
            const float inv  = 1.0f / rsum;
            float cost = 0.0f, lss = 0.0f;
            const float bv = beta_v[c];
#pragma unroll
            for (int d = 0; d < DD; ++d) {
                const float m  = s_accM[c * DD + d] * inv;
                float va = s_accV[c * DD + d] * inv - m * m;
                va = fmaxf(va, 0.0f);
                const float sd = sqrtf(va);
                s_mean [c * DD + d] = m;
                s_inv2v[c * DD + d] = 1.0f / (2.0f * va + EPSf);
                const float l = __logf(sd + EPSf);
                lss  += l;
                cost += bv + l;
            }
            cost *= rsum;
            const float inv_temp = 1.0f + (float)it;            // 1,2,3
            const float oact = 1.0f / (1.0f + __expf(-(inv_temp * (beta_a[c] - cost))));
            s_logstd[c] = lss;
            s_logact[c] = __logf(oact + EPSf);
            if (it == 2) {
                for (int d = 0; d < DD; ++d)
                    out[b * CD + c * DD + d] = s_mean[c * DD + d];
                out[Bsz * CD + b * Csz + c] = oact;
            }
        }
        __syncthreads();
    }
}

extern "C" void kernel_launch(void* const* d_in, const int* in_sizes, int n_in,
                              void* d_out, int out_size, void* d_ws, size_t ws_size,
                              hipStream_t stream) {
    const float* pose = (const float*)d_in[0];
    const float* act  = (const float*)d_in[1];
    const float* w    = (const float*)d_in[2];
    const float* bv   = (const float*)d_in[3];
    const float* ba   = (const float*)d_in[4];
    float* out = (float*)d_out;
    em_caps_kernel<<<dim3(Bsz), dim3(THREADS), 0, stream>>>(pose, act, w, bv, ba, out);
}